// GIN_49632642073100
// MI455X (gfx1250) — compile-verified
//
#include <hip/hip_runtime.h>

typedef __attribute__((ext_vector_type(2))) float v2f;
typedef __attribute__((ext_vector_type(8))) float v8f;

#define FEAT 64
static constexpr float BN_EPS_C = 1e-5f;

// ---------------------------------------------------------------------------
// Scatter-add over edges: agg[dst[e]][c] += bn(h[src[e]][c])
// One thread per (edge, 4-column group): vector b128 gather (coalesced 256B
// per edge across 16 threads) + 4 f32 atomics that resolve in L2
// (agg = 25.6MB << 192MB L2). Lazy BN (previous layer's outer BatchNorm)
// is applied as a compile-time-selected scale/shift on the gathered value.
// ---------------------------------------------------------------------------
template<bool HAS_BN>
__global__ void gin_scatter(const float* __restrict__ h,
                            const int* __restrict__ src,
                            const int* __restrict__ dst,
                            const float* __restrict__ sc,
                            const float* __restrict__ sh,
                            float* __restrict__ agg,
                            long long total)            // E * 16
{
    long long idx = (long long)blockIdx.x * blockDim.x + threadIdx.x;
    if (idx >= total) return;
    int e  = (int)(idx >> 4);
    int c0 = (int)(idx & 15) * 4;

    float4 v4 = *(const float4*)(h + (long long)src[e] * FEAT + c0);
    float vals[4] = {v4.x, v4.y, v4.z, v4.w};
    float* arow = agg + (long long)dst[e] * FEAT + c0;

#pragma unroll
    for (int j = 0; j < 4; ++j) {
        float v = vals[j];
        if (HAS_BN) v = fmaf(sc[c0 + j], v, sh[c0 + j]);
        atomicAdd(arow + j, v);
    }
}

// ---------------------------------------------------------------------------
// Fused GEMM using native fp32 WMMA (V_WMMA_F32_16X16X4_F32).
// One wave computes a 16-row strip x NTILES*16 columns, K = 64 (16 wmma steps
// per tile). A operand transform (lazy BN + optional ReLU + optional +agg) is
// fused into the load; epilogue fuses bias, optional ReLU, masked store and
// per-column sum/sumsq accumulation for the next BatchNorm. All mode flags
// are template parameters so the 16-step k-loop is branch-free.
//
// fp32 WMMA operand layout (ISA 7.12.2, 32-bit A 16x4):
//   A: lanes 0-15 -> rows m0..m0+15, VGPR0=K k+0, VGPR1=K k+1
//      lanes 16-31 -> same rows,     VGPR0=K k+2, VGPR1=K k+3
//   B: symmetric with N on lanes.
//   C/D: 8 VGPRs, col = (lane&15)+n0, row = v + 8*(lane>=16) + m0.
// ---------------------------------------------------------------------------
template<int NTILES, int NCOLS, bool HAS_BN, bool RELU_IN, bool HAS_AGG,
         bool EPI_RELU, bool EPI_STATS, bool MASKCOL>
__global__ void gin_gemm(const float* __restrict__ A,
                         const float* __restrict__ isc,   // input scale (HAS_BN)
                         const float* __restrict__ ish,   // input shift (HAS_BN)
                         const float* __restrict__ agg,   // aggregation (HAS_AGG)
                         const float* __restrict__ W,     // 64 x NCOLS, row major
                         const float* __restrict__ bias,  // NCOLS
                         float* __restrict__ out,         // nrows x NCOLS
                         float* __restrict__ statSum,
                         float* __restrict__ statSq,
                         int nrows)
{
    int lane  = threadIdx.x & 31;
    int wave  = threadIdx.x >> 5;
    int strip = blockIdx.x * (blockDim.x >> 5) + wave;
    int m0 = strip * 16;
    if (m0 >= nrows) return;
    int half = lane >> 4;     // which K-pair this lane supplies
    int lm   = lane & 15;

    int arow = m0 + lm;
    if (arow >= nrows) arow = nrows - 1;     // safe clamp (N%16==0 in practice)
    const float* Arow = A + (long long)arow * FEAT;
    const float* Grow = HAS_AGG ? agg + (long long)arow * FEAT : nullptr;

    v8f acc[NTILES] = {};

#pragma unroll
    for (int k = 0; k < FEAT; k += 4) {
        int k0 = k + 2 * half;                       // even -> 8B aligned
        v2f a = *(const v2f*)(Arow + k0);
        if (HAS_BN) {
            a.x = fmaf(isc[k0],     a.x, ish[k0]);
            a.y = fmaf(isc[k0 + 1], a.y, ish[k0 + 1]);
        }
        if (RELU_IN) { a.x = fmaxf(a.x, 0.f); a.y = fmaxf(a.y, 0.f); }
        if (HAS_AGG) { a.x += Grow[k0]; a.y += Grow[k0 + 1]; }

#pragma unroll
        for (int t = 0; t < NTILES; ++t) {
            int col = t * 16 + lm;
            v2f b;
            if (!MASKCOL || col < NCOLS) {
                b.x = W[k0 * NCOLS + col];
                b.y = W[(k0 + 1) * NCOLS + col];
            } else {
                b.x = 0.f; b.y = 0.f;
            }
            acc[t] = __builtin_amdgcn_wmma_f32_16x16x4_f32(
                false, a, false, b, (short)0, acc[t], false, false);
        }
    }

    // Epilogue: bias (+ReLU) (+masked) store, fused BN-stat accumulation.
#pragma unroll
    for (int t = 0; t < NTILES; ++t) {
        int col = t * 16 + lm;
        bool colok = (!MASKCOL) || (col < NCOLS);
        float bv = colok ? bias[col] : 0.f;
        float s = 0.f, sq = 0.f;
#pragma unroll
        for (int v = 0; v < 8; ++v) {
            int row = m0 + v + 8 * half;
            float y = acc[t][v] + bv;
            if (EPI_RELU) y = fmaxf(y, 0.f);
            if (colok && row < nrows)
                out[(long long)row * NCOLS + col] = y;
            if (EPI_STATS) { s += y; sq += y * y; }
        }
        if (EPI_STATS && colok) {
            atomicAdd(&statSum[col], s);
            atomicAdd(&statSq[col],  sq);
        }
    }
}

// ---------------------------------------------------------------------------
// Fold BN stats + gamma/beta into an affine (scale, shift) pair.
// ---------------------------------------------------------------------------
__global__ void gin_bn_finalize(const float* __restrict__ sum,
                                const float* __restrict__ sumsq,
                                const float* __restrict__ g,
                                const float* __restrict__ bt,
                                float* __restrict__ scale,
                                float* __restrict__ shift,
                                int ncols, float invN)
{
    int c = blockIdx.x * blockDim.x + threadIdx.x;
    if (c >= ncols) return;
    float mu  = sum[c] * invN;
    float var = sumsq[c] * invN - mu * mu;
    float sc  = g[c] * rsqrtf(var + BN_EPS_C);
    scale[c] = sc;
    shift[c] = fmaf(-mu, sc, bt[c]);
}

// ---------------------------------------------------------------------------
// Row-wise log_softmax over 40 columns: one wave32 per row, shfl reductions.
// ---------------------------------------------------------------------------
__global__ void gin_logsoftmax40(const float* __restrict__ Y,
                                 float* __restrict__ out, int nrows)
{
    int row = blockIdx.x * (blockDim.x >> 5) + (threadIdx.x >> 5);
    if (row >= nrows) return;
    int lane = threadIdx.x & 31;
    const float* yr = Y + (long long)row * 40;

    float v0 = yr[lane];                          // lanes 0..31 all < 40
    bool has1 = (lane + 32) < 40;                 // lanes 0..7
    float v1 = has1 ? yr[lane + 32] : -3.402823466e38f;

    float m = fmaxf(v0, v1);
#pragma unroll
    for (int off = 16; off > 0; off >>= 1)
        m = fmaxf(m, __shfl_xor(m, off, 32));

    float e = expf(v0 - m) + (has1 ? expf(v1 - m) : 0.f);
#pragma unroll
    for (int off = 16; off > 0; off >>= 1)
        e += __shfl_xor(e, off, 32);

    float lse = m + logf(e);
    out[(long long)row * 40 + lane] = v0 - lse;
    if (has1) out[(long long)row * 40 + lane + 32] = v1 - lse;
}

// ---------------------------------------------------------------------------
extern "C" void kernel_launch(void* const* d_in, const int* in_sizes, int n_in,
                              void* d_out, int out_size, void* d_ws, size_t ws_size,
                              hipStream_t stream)
{
    const float* x   = (const float*)d_in[0];
    const int*   src = (const int*)d_in[1];
    const int*   dst = (const int*)d_in[2];
    const float* W1[3] = {(const float*)d_in[3],  (const float*)d_in[9],  (const float*)d_in[15]};
    const float* b1[3] = {(const float*)d_in[4],  (const float*)d_in[10], (const float*)d_in[16]};
    const float* g [3] = {(const float*)d_in[5],  (const float*)d_in[11], (const float*)d_in[17]};
    const float* bt[3] = {(const float*)d_in[6],  (const float*)d_in[12], (const float*)d_in[18]};
    const float* W2[3] = {(const float*)d_in[7],  (const float*)d_in[13], (const float*)d_in[19]};
    const float* b2[3] = {(const float*)d_in[8],  (const float*)d_in[14], (const float*)d_in[20]};
    const float* bng[2] = {(const float*)d_in[21], (const float*)d_in[23]};
    const float* bnb[2] = {(const float*)d_in[22], (const float*)d_in[24]};

    int N = in_sizes[0] / FEAT;
    long long E = in_sizes[1];

    // Workspace carve-up (floats): agg | z | h | stats; final 40-col logits
    // reuse the agg region (its data is consumed before the last GEMM writes).
    float* AGG = (float*)d_ws;
    float* Z   = AGG + (long long)N * FEAT;
    float* U   = Z   + (long long)N * FEAT;
    float* ST  = U   + (long long)N * FEAT;
    float* sum1 = ST + 0,   *sq1 = ST + 64;
    float* sum2 = ST + 128, *sq2 = ST + 192;
    float* sc1  = ST + 256, *sh1 = ST + 320;
    float* scO[2] = {ST + 384, ST + 512};
    float* shO[2] = {ST + 448, ST + 576};
    float* Y2 = AGG;

    int strips = (N + 15) / 16;
    dim3 gBlk(128);                          // 4 waves/block
    dim3 gGrd((unsigned)((strips + 3) / 4));
    long long etotal = E * 16;               // (edge, 4-col group) work items
    dim3 sBlk(256), sGrd((unsigned)((etotal + 255) / 256));
    float invN = 1.0f / (float)N;

    for (int i = 0; i < 3; ++i) {
        const float* hin = (i == 0) ? x : U;
        const float* isc = (i == 0) ? nullptr : scO[i - 1];
        const float* ish = (i == 0) ? nullptr : shO[i - 1];

        hipMemsetAsync(AGG, 0, (size_t)N * FEAT * sizeof(float), stream);
        hipMemsetAsync(ST, 0, 256 * sizeof(float), stream);   // sum1,sq1,sum2,sq2

        if (i == 0) {
            gin_scatter<false><<<sGrd, sBlk, 0, stream>>>(hin, src, dst,
                                                          nullptr, nullptr, AGG, etotal);
            // z = (h + agg) @ W1 + b1 ; accumulate z column stats
            gin_gemm<4, 64, false, false, true, false, true, false>
                <<<gGrd, gBlk, 0, stream>>>(hin, nullptr, nullptr, AGG,
                                            W1[i], b1[i], Z, sum1, sq1, N);
        } else {
            gin_scatter<true><<<sGrd, sBlk, 0, stream>>>(hin, src, dst,
                                                         isc, ish, AGG, etotal);
            // z = (bn(h) + agg) @ W1 + b1 ; accumulate z column stats
            gin_gemm<4, 64, true, false, true, false, true, false>
                <<<gGrd, gBlk, 0, stream>>>(hin, isc, ish, AGG,
                                            W1[i], b1[i], Z, sum1, sq1, N);
        }

        gin_bn_finalize<<<1, 64, 0, stream>>>(sum1, sq1, g[i], bt[i], sc1, sh1, 64, invN);

        if (i < 2) {
            // u = relu( relu(bn(z)) @ W2 + b2 ) ; accumulate u stats for outer BN
            gin_gemm<4, 64, true, true, false, true, true, false>
                <<<gGrd, gBlk, 0, stream>>>(Z, sc1, sh1, nullptr,
                                            W2[i], b2[i], U, sum2, sq2, N);
            gin_bn_finalize<<<1, 64, 0, stream>>>(sum2, sq2, bng[i], bnb[i],
                                                  scO[i], shO[i], 64, invN);
        } else {
            // final logits: 40 columns, masked stores, no stats
            gin_gemm<3, 40, true, true, false, false, false, true>
                <<<gGrd, gBlk, 0, stream>>>(Z, sc1, sh1, nullptr,
                                            W2[i], b2[i], Y2, nullptr, nullptr, N);
        }
    }

    gin_logsoftmax40<<<(unsigned)((N + 7) / 8), 256, 0, stream>>>(Y2, (float*)d_out, N);
}